// TopographicalCorticalSheet_24300924961002
// MI455X (gfx1250) — compile-verified
//
#include <hip/hip_runtime.h>
#include <stdint.h>

typedef float v2f __attribute__((ext_vector_type(2)));
typedef float v8f __attribute__((ext_vector_type(8)));

#define NN       65536     // neurons (256*256)
#define KP1      33        // synapses per neuron + self
#define BATCH    64        // batch size
#define CPB      8         // columns per block (1 column per wave32)
#define NTHREADS (CPB * 32)

__global__ __launch_bounds__(NTHREADS)
void sheet_zero(float4* __restrict__ out, int n4) {
  int i = blockIdx.x * blockDim.x + threadIdx.x;
  if (i < n4) out[i] = make_float4(0.f, 0.f, 0.f, 0.f);
}

// out[row[e], :] += vals[e] * x[e/33, :]   (scatter-SpMM, CSC with implicit col)
__global__ __launch_bounds__(NTHREADS)
void sheet_scatter_spmm(const float* __restrict__ x,
                        const float* __restrict__ vals,
                        const unsigned* __restrict__ row,
                        float*       __restrict__ out)
{
  __shared__ __align__(16) unsigned s_row[CPB * KP1];  // 264 u32 = 1056 B (66 x 16B)
  const int tid  = threadIdx.x;
  const int lane = tid & 31;
  const int wv   = tid >> 5;            // wave id in block = column slot
  const int jblk = blockIdx.x * CPB;
  const int j    = jblk + wv;           // this wave's column (neuron)

  // ---- async stage of this block's row indices: 264 contiguous u32 -> LDS ----
  if (tid < (CPB * KP1) / 4) {          // 66 lanes copy 16B each
    const unsigned* gsrc = row + (size_t)jblk * KP1 + tid * 4;
    unsigned lds_off = (unsigned)(uintptr_t)&s_row[tid * 4];  // low 32 bits = LDS offset
    asm volatile("global_load_async_to_lds_b128 %0, %1, off"
                 :: "v"(lds_off), "v"(gsrc) : "memory");
  }
  asm volatile("s_wait_asynccnt 0" ::: "memory");
  __syncthreads();

  const size_t   ebase = (size_t)j * KP1;
  const unsigned hh    = (unsigned)lane >> 4;   // half-wave: 0 or 1
  const unsigned l16   = (unsigned)lane & 15;

  // ---- slots 0..31 via WMMA rank-1 outer products (2 groups of 16 edges) ----
#pragma unroll
  for (int g = 0; g < 2; ++g) {
    // A[16x4] layout: lanes 0-15 hold {K=0,K=1}, lanes 16-31 hold {K=2,K=3}.
    // One-hot K=0: A[M,0] = vals[edge M]; all K>=1 of A are zero.
    const unsigned aslot = g * 16 + l16;               // <= 31, always in range
    const float avl = vals[ebase + aslot];
    v2f Av; Av.x = (hh == 0) ? avl : 0.f; Av.y = 0.f;

#pragma unroll
    for (int cb = 0; cb < 4; ++cb) {    // 4 batch chunks of 16
      // B[4x16]: only the K=0 row matters (A's zeros kill K>=1): x[j, b0..b0+15]
      const float bxl = x[(size_t)j * BATCH + cb * 16 + l16];
      v2f Bv; Bv.x = (hh == 0) ? bxl : 0.f; Bv.y = 0.f;

      v8f C = {0.f, 0.f, 0.f, 0.f, 0.f, 0.f, 0.f, 0.f};
      // D[M,N] = vals[edge M] * x[j, b0+N]
      v8f D = __builtin_amdgcn_wmma_f32_16x16x4_f32(
                  false, Av, false, Bv, (short)0, C, false, false);

      // D layout: VGPR p, lanes 0-15 -> M=p ; lanes 16-31 -> M=p+8
#pragma unroll
      for (int p = 0; p < 8; ++p) {
        const unsigned slot = g * 16 + p + hh * 8;     // <= 31, no guard needed
        const unsigned r    = s_row[wv * KP1 + slot];
        const unsigned off  = (r << 6) + cb * 16 + l16;  // 32-bit elem offset
        unsafeAtomicAdd(out + (size_t)off, D[p]);
      }
    }
  }

  // ---- slot 32 (appended self-edge) via plain VALU: 2 batch elems per lane ----
  {
    const unsigned r32 = s_row[wv * KP1 + 32];
    const float    v32 = vals[ebase + 32];
    const float2   x2  = ((const float2*)x)[((size_t)j * BATCH >> 1) + lane];
    const unsigned off = (r32 << 6) + 2 * lane;
    unsafeAtomicAdd(out + (size_t)off,     v32 * x2.x);
    unsafeAtomicAdd(out + (size_t)off + 1, v32 * x2.y);
  }
}

extern "C" void kernel_launch(void* const* d_in, const int* in_sizes, int n_in,
                              void* d_out, int out_size, void* d_ws, size_t ws_size,
                              hipStream_t stream) {
  const float*    x    = (const float*)d_in[0];     // [N, B] f32
  const float*    vals = (const float*)d_in[1];     // [E] f32
  const unsigned* row  = (const unsigned*)d_in[2];  // [E] i32 indices in [0, N)
  // d_in[3] = col is structured: col[e] == e / 33, used implicitly.
  float* out = (float*)d_out;                       // [N, B] f32

  const int n4 = out_size / 4;                      // 1,048,576 float4
  sheet_zero<<<(n4 + NTHREADS - 1) / NTHREADS, NTHREADS, 0, stream>>>(
      (float4*)out, n4);
  sheet_scatter_spmm<<<NN / CPB, NTHREADS, 0, stream>>>(x, vals, row, out);
}